// mynetwork_44590350467201
// MI455X (gfx1250) — compile-verified
//
#include <hip/hip_runtime.h>
#include <math.h>

typedef float v2f __attribute__((ext_vector_type(2)));
typedef float v8f __attribute__((ext_vector_type(8)));

#define NUM_RFS 8192
#define DIM     4096
#define BLOCK   256
#define ROWS_PER_GROUP 16
#define GROUPS_PER_BLOCK 4          // 4 groups x 2 K-halves = 8 waves
#define ROWS_PER_BLOCK  (ROWS_PER_GROUP * GROUPS_PER_BLOCK)   // 64
#define KSPLIT          2
#define KHALF           (DIM / KSPLIT)                        // 2048
#define BLOCKS_PER_MAT  (NUM_RFS / ROWS_PER_BLOCK)            // 128
#define INV_SQRT_NUM_RFS 0.011048543456039806f  // 1/sqrt(8192)

// ---------------------------------------------------------------------------
// Fused kernel: streaming WMMA GEMV + Performer feature map, both matrices.
//   Grid = 256 blocks: blocks [0,128) -> gs_x, [128,256) -> gs_w.
//   Block = 8 wave32: wave w -> row-group (w&3), K-half (w>>2).
//   Per wave, V_WMMA_F32_16X16X4_F32 over its K-half:
//     A (16x4):  lane L<16 -> gs[m0+L][k+0..1], lane L>=16 -> gs[m0+L-16][k+2..3]
//     B (4x16):  every column = vec[k..k+3] (broadcast from LDS)
//     C (16x16): all 16 columns identical; lane0 VGPR r = dot(row m0+r),
//                lane16 VGPR r = dot(row m0+8+r)
//   K-halves combined through LDS; block recomputes 0.5*||vec||^2 locally.
//   Epilogue: out[row] = exp(xi*dot - s*xi*xi) / sqrt(NUM_RFS)
// ---------------------------------------------------------------------------
__global__ void __launch_bounds__(BLOCK)
gemv_exp_wmma(const float* __restrict__ gs_x, const float* __restrict__ gs_w,
              const float* __restrict__ x,    const float* __restrict__ w,
              const float* __restrict__ xis,
              float* __restrict__ xb,         float* __restrict__ wb) {
    __shared__ float xs[DIM];                         // 16 KB vector cache
    __shared__ float part[GROUPS_PER_BLOCK][32][8];   // 4 KB K-half partials
    __shared__ float red[BLOCK];                      // 1 KB norm reduction
    __shared__ float snorm_sh;

    const int mat = (int)blockIdx.x >> 7;             // 0: x-path, 1: w-path
    const int blk = (int)blockIdx.x & (BLOCKS_PER_MAT - 1);
    const float* __restrict__ gs  = mat ? gs_w : gs_x;
    const float* __restrict__ vec = mat ? w : x;
    float* __restrict__ out       = mat ? wb : xb;

    const int t = threadIdx.x;
    for (int i = t; i < DIM; i += BLOCK) xs[i] = vec[i];
    __syncthreads();

    // Block-local 0.5*||vec||^2 (redundant per block; negligible vs 1MB stream)
    {
        float a = 0.f;
        for (int i = t; i < DIM; i += BLOCK) { float v = xs[i]; a += v * v; }
        red[t] = a; __syncthreads();
        for (int off = BLOCK / 2; off > 0; off >>= 1) {
            if (t < off) red[t] += red[t + off];
            __syncthreads();
        }
        if (t == 0) snorm_sh = 0.5f * red[0];
        __syncthreads();
    }

    const int wave  = t >> 5;
    const int lane  = t & 31;
    const int g     = wave & (GROUPS_PER_BLOCK - 1);  // row group in block
    const int khalf = wave >> 2;                      // 0 or 1
    const int m0    = blk * ROWS_PER_BLOCK + g * ROWS_PER_GROUP;
    const int k0    = khalf * KHALF;

    const int half = lane >> 4;                       // low/high half-wave
    const int koff = half * 2;                        // A/B K split {0,1}/{2,3}
    const float* __restrict__ arow =
        gs + (size_t)(m0 + (lane & 15)) * DIM + k0 + koff;

    v8f acc = {0.f, 0.f, 0.f, 0.f, 0.f, 0.f, 0.f, 0.f};

    #pragma unroll 16
    for (int k = 0; k < KHALF; k += 4) {
        // 268 MB streamed once (> 192 MB L2): nontemporal on the gs path.
        v2f a = __builtin_nontemporal_load((const v2f*)(arow + k));
        v2f b = *(const v2f*)(&xs[k0 + k + koff]);    // broadcast chunk (LDS)
        acc = __builtin_amdgcn_wmma_f32_16x16x4_f32(
            /*neg_a=*/false, a, /*neg_b=*/false, b,
            /*c_mod=*/(short)0, acc, /*reuse_a=*/false, /*reuse_b=*/false);
    }

    // Combine the two K-halves of each row group through LDS.
    if (khalf == 1) {
        #pragma unroll
        for (int r = 0; r < 8; ++r) part[g][lane][r] = acc[r];
    }
    __syncthreads();

    if (khalf == 0) {
        #pragma unroll
        for (int r = 0; r < 8; ++r) acc[r] += part[g][lane][r];

        // Lanes 0 and 16 each own 8 distinct row results (C-matrix layout).
        if ((lane & 15) == 0) {
            const float s = snorm_sh;
            #pragma unroll
            for (int r = 0; r < 8; ++r) {
                const int row = m0 + half * 8 + r;
                const float xi = xis[row];
                const float d  = acc[r];
                out[row] = INV_SQRT_NUM_RFS * expf(xi * d - s * xi * xi);
            }
        }
    }
}

// ---------------------------------------------------------------------------
// Final: scalar = xb . wb
// ---------------------------------------------------------------------------
__global__ void final_dot(const float* __restrict__ xb,
                          const float* __restrict__ wb,
                          float* __restrict__ outp) {
    __shared__ float red[BLOCK];
    const int t = threadIdx.x;
    float a = 0.f;
    for (int i = t; i < NUM_RFS; i += BLOCK) a += xb[i] * wb[i];
    red[t] = a; __syncthreads();
    for (int off = BLOCK / 2; off > 0; off >>= 1) {
        if (t < off) red[t] += red[t + off];
        __syncthreads();
    }
    if (t == 0) outp[0] = red[0];
}

// ---------------------------------------------------------------------------
// Launch
// ---------------------------------------------------------------------------
extern "C" void kernel_launch(void* const* d_in, const int* in_sizes, int n_in,
                              void* d_out, int out_size, void* d_ws, size_t ws_size,
                              hipStream_t stream) {
    const float* x    = (const float*)d_in[0];   // [DIM]
    const float* w    = (const float*)d_in[1];   // [DIM]
    const float* xis  = (const float*)d_in[2];   // [NUM_RFS]
    const float* gs_x = (const float*)d_in[3];   // [NUM_RFS, DIM]
    const float* gs_w = (const float*)d_in[4];   // [NUM_RFS, DIM]
    float* outp = (float*)d_out;

    float* xb = (float*)d_ws;                    // [NUM_RFS]
    float* wb = xb + NUM_RFS;                    // [NUM_RFS]

    gemv_exp_wmma<<<2 * BLOCKS_PER_MAT, BLOCK, 0, stream>>>(
        gs_x, gs_w, x, w, xis, xb, wb);

    final_dot<<<1, BLOCK, 0, stream>>>(xb, wb, outp);
}